// DPLQR_74122545594829
// MI455X (gfx1250) — compile-verified
//
#include <hip/hip_runtime.h>

typedef __attribute__((ext_vector_type(2))) float v2f;
typedef __attribute__((ext_vector_type(8))) float v8f;
typedef __attribute__((ext_vector_type(4))) int v4i;

constexpr int NS = 32;    // N_STATE
constexpr int NC = 16;    // N_CTRL
constexpr int NSC = 48;   // NS + NC
constexpr int TT = 128;   // horizon
constexpr int NBATCH = 64;
constexpr int LDQ = NSC + 1;  // padded leading dim (odd -> conflict-free cols)

// ---- async global->LDS staging (GLOBAL_LOAD_ASYNC_TO_LDS_B128, ASYNCcnt) ----
#if defined(__HIP_DEVICE_COMPILE__) && defined(__has_builtin)
#if __has_builtin(__builtin_amdgcn_global_load_async_to_lds_b128)
#define USE_ASYNC_LDS 1
#endif
#endif
#ifndef USE_ASYNC_LDS
#define USE_ASYNC_LDS 0
#endif

// Stage one 48x48 f32 Q slab into padded LDS buffer (48 rows x 3 x 16B chunks).
__device__ __forceinline__ void q_stage_issue(const float* __restrict__ Qr,
                                              float (*dst)[LDQ], int tid) {
#if USE_ASYNC_LDS
  typedef __attribute__((address_space(1))) v4i* gp;
  typedef __attribute__((address_space(3))) v4i* lp;
  for (int c = tid; c < NSC * 3; c += 256) {
    const int row = c / 3, ch = (c % 3) * 4;
    __builtin_amdgcn_global_load_async_to_lds_b128((gp)(Qr + row * NSC + ch),
                                                   (lp)&dst[row][ch], 0, 0);
  }
#else
  for (int idx = tid; idx < NSC * NSC; idx += 256)
    dst[idx / NSC][idx % NSC] = Qr[idx];
#endif
}

__device__ __forceinline__ void q_stage_wait() {
#if USE_ASYNC_LDS
#if __has_builtin(__builtin_amdgcn_s_wait_asynccnt)
  __builtin_amdgcn_s_wait_asynccnt(0);
#else
  asm volatile("s_wait_asynccnt 0" ::: "memory");
#endif
#endif
}

// Block-cooperative tiled f32 GEMM: D = (opA)*B [+ Cin], 16x16 tiles,
// K consumed 4 at a time via V_WMMA_F32_16X16X4_F32. Wave-uniform tile loop
// keeps EXEC all-ones for WMMA. Row-major operands with given ld.
template <bool A_TRANS, bool HAVE_C, bool NEGATE>
__device__ __forceinline__ void mm_tiles(const float* __restrict__ A, int lda,
                                         const float* __restrict__ B, int ldb,
                                         const float* __restrict__ Cin, int ldc,
                                         float* __restrict__ D, int ldd,
                                         int M, int N, int K,
                                         int wave, int nwaves, int lane) {
  const int mt = M >> 4, nt = N >> 4, nk = K >> 2;
  const int n16 = lane & 15;           // M-index (A) / N-index (B,C,D)
  const int khalf = (lane >> 4) << 1;  // lanes 16-31 hold K = 2,3 of each K=4 slab
  const int mbase = (lane >> 4) << 3;  // C/D: lanes 16-31 hold M = r+8
  for (int tile = wave; tile < mt * nt; tile += nwaves) {
    const int ti = tile / nt, tj = tile % nt;
    const int r0 = ti << 4, c0 = tj << 4;
    v8f acc = {};
    if (HAVE_C) {
#pragma unroll
      for (int r = 0; r < 8; ++r)
        acc[r] = Cin[(size_t)(r0 + mbase + r) * ldc + c0 + n16];
    }
    for (int kk = 0; kk < nk; ++kk) {
      const int k0 = kk << 2;
      v2f a, b;
      if (A_TRANS) {  // logical A[m][k] = Aphys[k][m]
        a.x = A[(k0 + khalf) * lda + r0 + n16];
        a.y = A[(k0 + khalf + 1) * lda + r0 + n16];
      } else {
        a.x = A[(r0 + n16) * lda + k0 + khalf];
        a.y = A[(r0 + n16) * lda + k0 + khalf + 1];
      }
      b.x = B[(k0 + khalf) * ldb + c0 + n16];
      b.y = B[(k0 + khalf + 1) * ldb + c0 + n16];
      acc = __builtin_amdgcn_wmma_f32_16x16x4_f32(false, a, false, b,
                                                  (short)0, acc, false, false);
    }
#pragma unroll
    for (int r = 0; r < 8; ++r) {
      float v = NEGATE ? -acc[r] : acc[r];
      D[(size_t)(r0 + mbase + r) * ldd + c0 + n16] = v;
    }
  }
}

__global__ __launch_bounds__(256) void lqr_fused_kernel(
    const float* __restrict__ x_init, const float* __restrict__ Qg,
    const float* __restrict__ pg, const float* __restrict__ Ag,
    const float* __restrict__ Bg, const float* __restrict__ c1g,
    float* __restrict__ out, float* __restrict__ Kws, float* __restrict__ kws) {
  const int b = blockIdx.x;
  const int tid = threadIdx.x;
  const int lane = tid & 31;
  const int wave = tid >> 5;
  const int nwaves = 8;

  __shared__ float Qraw[2][NSC][LDQ];  // double-buffered Q[b][t] staging
  __shared__ float Fs[NS][LDQ];        // F = [A | B]
  __shared__ float Qs[NSC][LDQ];       // accumulated Qt (t < T-1)
  __shared__ float M1[NS][LDQ];        // V @ F
  __shared__ float Vs[NS][NS + 1];     // value matrix V
  __shared__ float Ks[NC][NS + 1];     // gain K
  __shared__ float T2s[NC][NS + 1];    // Qux + Quu K
  __shared__ float GJ[NC][2 * NC + 1]; // Gauss-Jordan [Quu | I]
  __shared__ float fac[NC];
  __shared__ float qs[NSC], wv[NS], zz[NC], kv[NC], ftv[NS], vv[NS];
  __shared__ float xcur[NS], xnew[NS], uu[NC], xu[NSC], red[NSC];

  // Build F and ft = c1 (persist for whole kernel); stage Q[T-1].
  q_stage_issue(Qg + ((size_t)b * TT + TT - 1) * NSC * NSC, Qraw[(TT - 1) & 1],
                tid);
  for (int idx = tid; idx < NS * NSC; idx += 256) {
    int i = idx / NSC, j = idx % NSC;
    Fs[i][j] = (j < NS) ? Ag[((size_t)b * NS + i) * NS + j]
                        : Bg[((size_t)b * NS + i) * NC + (j - NS)];
  }
  if (tid < NS) ftv[tid] = c1g[(size_t)b * NS + tid];
  __syncthreads();

  // ---------------- backward Riccati recursion ----------------
  for (int t = TT - 1; t >= 0; --t) {
    const int qb = t & 1;
    const float* pr = pg + ((size_t)b * TT + t) * NSC;

    q_stage_wait();   // this wave's staged chunks of Q[t] complete
    __syncthreads();  // all waves' chunks visible
    if (t > 0)        // overlap next slab's DMA with this step's compute
      q_stage_issue(Qg + ((size_t)b * TT + t - 1) * NSC * NSC, Qraw[qb ^ 1],
                    tid);

    const float* Qc;  // current Qt, leading dim LDQ
    if (t == TT - 1) {
      Qc = &Qraw[qb][0][0];
      if (tid < NSC) qs[tid] = pr[tid];
      __syncthreads();
    } else {
      Qc = &Qs[0][0];
      // w = V ft + v
      if (tid < NS) {
        float s = vv[tid];
        for (int j = 0; j < NS; ++j) s += Vs[tid][j] * ftv[j];
        wv[tid] = s;
      }
      __syncthreads();
      // M1 = V @ F  (32x48, K=32)
      mm_tiles<false, false, false>(&Vs[0][0], NS + 1, &Fs[0][0], LDQ, nullptr,
                                    0, &M1[0][0], LDQ, NS, NSC, NS, wave,
                                    nwaves, lane);
      __syncthreads();
      // Qt = Qraw + F^T @ M1  (48x48, K=32), C sourced from staged LDS slab
      mm_tiles<true, true, false>(&Fs[0][0], LDQ, &M1[0][0], LDQ,
                                  &Qraw[qb][0][0], LDQ, &Qs[0][0], LDQ, NSC,
                                  NSC, NS, wave, nwaves, lane);
      // qt = p + F^T w
      if (tid < NSC) {
        float s = pr[tid];
        for (int k = 0; k < NS; ++k) s += Fs[k][tid] * wv[k];
        qs[tid] = s;
      }
      __syncthreads();
    }

    // Gauss-Jordan inverse of Quu (SPD -> diagonal pivots OK)
    for (int idx = tid; idx < NC * 2 * NC; idx += 256) {
      int i = idx / (2 * NC), j = idx % (2 * NC);
      GJ[i][j] = (j < NC) ? Qc[(NS + i) * LDQ + NS + j]
                          : ((j - NC == i) ? 1.f : 0.f);
    }
    __syncthreads();
    for (int piv = 0; piv < NC; ++piv) {
      float pd = 1.0f / GJ[piv][piv];
      if (tid < NC) fac[tid] = GJ[tid][piv];
      __syncthreads();
      if (tid < 2 * NC) GJ[piv][tid] *= pd;
      __syncthreads();
      for (int idx = tid; idx < NC * 2 * NC; idx += 256) {
        int i = idx / (2 * NC), j = idx % (2 * NC);
        if (i != piv) GJ[i][j] -= fac[i] * GJ[piv][j];
      }
      __syncthreads();
    }

    // K = -(Quu_inv @ Qux)  (16x32, K=16)
    mm_tiles<false, false, true>(&GJ[0][NC], 2 * NC + 1, Qc + NS * LDQ, LDQ,
                                 nullptr, 0, &Ks[0][0], NS + 1, NC, NS, NC,
                                 wave, nwaves, lane);
    if (tid < NC) {
      float s = 0.f;
      for (int j = 0; j < NC; ++j) s += GJ[tid][NC + j] * qs[NS + j];
      kv[tid] = -s;
    }
    __syncthreads();
    // T2 = Qux + Quu @ K
    mm_tiles<false, true, false>(Qc + NS * LDQ + NS, LDQ, &Ks[0][0], NS + 1,
                                 Qc + NS * LDQ, LDQ, &T2s[0][0], NS + 1, NC, NS,
                                 NC, wave, nwaves, lane);
    if (tid < NC) {
      float s = qs[NS + tid];
      for (int j = 0; j < NC; ++j) s += Qc[(NS + tid) * LDQ + NS + j] * kv[j];
      zz[tid] = s;
    }
    __syncthreads();
    // V = Qxx + Qxu@K, then V += K^T@T2 (same wave owns same tiles)
    mm_tiles<false, true, false>(Qc + NS, LDQ, &Ks[0][0], NS + 1, Qc, LDQ,
                                 &Vs[0][0], NS + 1, NS, NS, NC, wave, nwaves,
                                 lane);
    mm_tiles<true, true, false>(&Ks[0][0], NS + 1, &T2s[0][0], NS + 1,
                                &Vs[0][0], NS + 1, &Vs[0][0], NS + 1, NS, NS,
                                NC, wave, nwaves, lane);
    // v = qx + Qxu k + K^T (qu + Quu k)
    if (tid < NS) {
      float s = qs[tid];
      for (int j = 0; j < NC; ++j)
        s += Qc[tid * LDQ + NS + j] * kv[j] + Ks[j][tid] * zz[j];
      vv[tid] = s;
    }
    // persist gains for forward pass
    {
      float* Kt = Kws + ((size_t)t * NBATCH + b) * NC * NS;
      for (int idx = tid; idx < NC * NS; idx += 256)
        Kt[idx] = Ks[idx / NS][idx % NS];
      if (tid < NC) kws[((size_t)t * NBATCH + b) * NC + tid] = kv[tid];
    }
    __syncthreads();
  }

  // ---------------- forward rollout ----------------
  float* xs_o = out;
  float* us_o = out + (size_t)TT * NBATCH * NS;
  float* objs_o = us_o + (size_t)TT * NBATCH * NC;
  float* taus_o = objs_o + (size_t)TT * NBATCH;

  q_stage_issue(Qg + (size_t)b * TT * NSC * NSC, Qraw[0], tid);
  if (tid < NS) xcur[tid] = x_init[(size_t)b * NS + tid];
  __syncthreads();

  for (int t = 0; t < TT; ++t) {
    const int qb = t & 1;
    const float* Kt = Kws + ((size_t)t * NBATCH + b) * NC * NS;
    const float* kt = kws + ((size_t)t * NBATCH + b) * NC;
    const float* pr = pg + ((size_t)b * TT + t) * NSC;

    q_stage_wait();
    __syncthreads();  // Qraw[qb] ready; prior iter's reads of qb^1 done
    if (t + 1 < TT)
      q_stage_issue(Qg + ((size_t)b * TT + t + 1) * NSC * NSC, Qraw[qb ^ 1],
                    tid);

    if (tid < NC) {
      float s = kt[tid];
      for (int j = 0; j < NS; ++j) s += Kt[tid * NS + j] * xcur[j];
      uu[tid] = s;
    }
    __syncthreads();
    if (tid < NSC) xu[tid] = (tid < NS) ? xcur[tid] : uu[tid - NS];
    __syncthreads();
    if (tid < NSC) {  // obj partials: xu_i * (0.5 * (Q xu)_i + p_i)
      float s = 0.f;
      for (int j = 0; j < NSC; ++j) s += Qraw[qb][tid][j] * xu[j];
      red[tid] = xu[tid] * (0.5f * s + pr[tid]);
    }
    if (tid >= 64 && tid < 64 + NS) {  // xn = A x + B u + c1
      int i = tid - 64;
      float s = ftv[i];
      for (int j = 0; j < NS; ++j) s += Fs[i][j] * xcur[j];
      for (int j = 0; j < NC; ++j) s += Fs[i][NS + j] * uu[j];
      xnew[i] = s;
    }
    __syncthreads();
    if (tid == 0) {
      float s = 0.f;
      for (int i = 0; i < NSC; ++i) s += red[i];
      objs_o[(size_t)t * NBATCH + b] = s;
    }
    if (tid < NS) xs_o[((size_t)t * NBATCH + b) * NS + tid] = xcur[tid];
    if (tid < NC) us_o[((size_t)t * NBATCH + b) * NC + tid] = uu[tid];
    if (tid < NSC) taus_o[((size_t)t * NBATCH + b) * NSC + tid] = xu[tid];
    __syncthreads();
    if (tid < NS) xcur[tid] = xnew[tid];
    __syncthreads();
  }
}

extern "C" void kernel_launch(void* const* d_in, const int* in_sizes, int n_in,
                              void* d_out, int out_size, void* d_ws,
                              size_t ws_size, hipStream_t stream) {
  const float* x_init = (const float*)d_in[0];
  const float* Q = (const float*)d_in[1];
  const float* p = (const float*)d_in[2];
  const float* A = (const float*)d_in[3];
  const float* Bm = (const float*)d_in[4];
  const float* c1 = (const float*)d_in[5];
  float* out = (float*)d_out;
  float* Kws = (float*)d_ws;                         // T*NB*NC*NS floats (16 MB)
  float* kws = Kws + (size_t)TT * NBATCH * NC * NS;  // T*NB*NC floats (0.5 MB)
  lqr_fused_kernel<<<NBATCH, 256, 0, stream>>>(x_init, Q, p, A, Bm, c1, out,
                                               Kws, kws);
}